// DeformConv2dPack_30863634989682
// MI455X (gfx1250) — compile-verified
//
#include <hip/hip_runtime.h>
#include <math.h>

typedef float v2f __attribute__((ext_vector_type(2)));
typedef float v8f __attribute__((ext_vector_type(8)));

namespace {

constexpr int Bn = 4, Cin = 64, Hh = 128, Ww = 128;
constexpr int Cout = 64;
constexpr int RED = Cin * 9;        // 576 = reduction length (c*9 taps)
constexpr int NPIX = Hh * Ww;

// ---------------- K1: NCHW -> NHWC transpose (32x32 LDS tiles) -------------
__global__ __launch_bounds__(256) void k_transpose(const float* __restrict__ x,
                                                   float* __restrict__ xt) {
  __shared__ float t[32][33];
  int b = blockIdx.z, h = blockIdx.y;
  int wt = blockIdx.x & 3;                 // W/32 = 4 tiles
  int ct = blockIdx.x >> 2;                // C/32 = 2 tiles
  int tx = threadIdx.x & 31, ty = threadIdx.x >> 5;   // 32 x 8
#pragma unroll
  for (int j = 0; j < 32; j += 8) {
    int c = ct * 32 + ty + j, w = wt * 32 + tx;
    t[ty + j][tx] = x[((b * Cin + c) * Hh + h) * Ww + w];   // coalesced in w
  }
  __syncthreads();
#pragma unroll
  for (int j = 0; j < 32; j += 8) {
    int w = wt * 32 + ty + j, c = ct * 32 + tx;
    xt[((b * Hh + h) * Ww + w) * Cin + c] = t[tx][ty + j];  // coalesced in c
  }
}

// ---------------- K2: offset conv 64 -> 18, 3x3, pad 1 ---------------------
// One block = 16x16 output pixels of one batch image. 18x18x64 halo in LDS.
__global__ __launch_bounds__(256) void k_offset_conv(
    const float* __restrict__ xt, const float* __restrict__ ow,
    const float* __restrict__ ob, float* __restrict__ offs) {
  constexpr int TP = 68;                       // padded channel stride / pixel
  __shared__ float tile[324 * TP];             // 18*18 pixels, 88 KB
  int b = blockIdx.y;
  int ht = blockIdx.x >> 3, wt = blockIdx.x & 7;
  int h0 = ht * 16, w0 = wt * 16;
  int tid = threadIdx.x;

  for (int i = tid; i < 324 * 16; i += 256) {  // float4 halo load
    int p = i >> 4, c4 = (i & 15) << 2;
    int gh = h0 - 1 + p / 18, gw = w0 - 1 + p % 18;
    float4 v = make_float4(0.f, 0.f, 0.f, 0.f);
    if (gh >= 0 && gh < Hh && gw >= 0 && gw < Ww)
      v = *(const float4*)(xt + ((b * Hh + gh) * Ww + gw) * Cin + c4);
    *(float4*)(tile + p * TP + c4) = v;
  }
  __syncthreads();

  int py = tid >> 4, px = tid & 15;
  int h = h0 + py, w = w0 + px;
  float acc[18];
#pragma unroll
  for (int oc = 0; oc < 18; ++oc) acc[oc] = ob[oc];
  for (int kk = 0; kk < 9; ++kk) {
    int ky = kk / 3, kx = kk % 3;
    int base = ((py + ky) * 18 + (px + kx)) * TP;
    for (int c = 0; c < 64; ++c) {
      float xv = tile[base + c];
#pragma unroll
      for (int oc = 0; oc < 18; ++oc)        // ow index uniform -> s_load
        acc[oc] = fmaf(xv, ow[oc * RED + c * 9 + kk], acc[oc]);
    }
  }
  float* op = offs + ((b * Hh + h) * Ww + w) * 18;   // pixel-major offsets
#pragma unroll
  for (int oc = 0; oc < 18; ++oc) op[oc] = acc[oc];
}

// ------------- K2b: reorder deform weights (o,c,k) -> (o, k*64+c) ----------
__global__ __launch_bounds__(256) void k_reorder_w(const float* __restrict__ dw,
                                                   float* __restrict__ dwr) {
  int i = blockIdx.x * 256 + threadIdx.x;
  if (i >= Cout * RED) return;
  int o = i / RED, r = i - o * RED;
  int k = r >> 6, c = r & 63;
  dwr[i] = dw[o * RED + c * 9 + k];
}

// ------------- K3: deformable gather + WMMA f32 GEMM -----------------------
// Block = 64 (COUT) x 32 (pixels along one row). 8 waves, one 16x16 C tile
// each, K=576 via v_wmma_f32_16x16x4_f32.
__global__ __launch_bounds__(256) void k_deform_wmma(
    const float* __restrict__ xt, const float* __restrict__ offs,
    const float* __restrict__ dwr, const float* __restrict__ db,
    float* __restrict__ out) {
  __shared__ float Bs[RED * 32];               // 576 x 32 im2col patch, 72 KB
  int b = blockIdx.y;
  int wt = blockIdx.x & 3;                     // W/32 = 4
  int h = blockIdx.x >> 2;
  int w0 = wt * 32;
  int tid = threadIdx.x;

  // --- build patch: task = (tap k, pixel n); lanes own distinct n -> LDS
  // column writes are bank-conflict free; each corner = 256B contiguous.
  for (int t = tid; t < 9 * 32; t += 256) {
    int k = t >> 5, n = t & 31;
    int w = w0 + n;
    const float* op = offs + (((b * Hh + h) * Ww + w) * 18 + 2 * k);
    float dy = op[0], dx = op[1];
    int ky = k / 3, kx = k % 3;
    float ys = (float)(h - 1 + ky) + dy;
    float xs = (float)(w - 1 + kx) + dx;
    float y0f = floorf(ys), x0f = floorf(xs);
    float ly = ys - y0f, lx = xs - x0f;
    int y0 = (int)y0f, x0 = (int)x0f;
    int y1 = y0 + 1, x1 = x0 + 1;
    float vy0 = (y0 >= 0 && y0 < Hh) ? 1.f : 0.f;
    float vy1 = (y1 >= 0 && y1 < Hh) ? 1.f : 0.f;
    float vx0 = (x0 >= 0 && x0 < Ww) ? 1.f : 0.f;
    float vx1 = (x1 >= 0 && x1 < Ww) ? 1.f : 0.f;
    float w00 = (1.f - ly) * (1.f - lx) * vy0 * vx0;
    float w01 = (1.f - ly) * lx * vy0 * vx1;
    float w10 = ly * (1.f - lx) * vy1 * vx0;
    float w11 = ly * lx * vy1 * vx1;
    int y0c = min(max(y0, 0), Hh - 1), y1c = min(max(y1, 0), Hh - 1);
    int x0c = min(max(x0, 0), Ww - 1), x1c = min(max(x1, 0), Ww - 1);
    const float* p00 = xt + ((b * Hh + y0c) * Ww + x0c) * Cin;
    const float* p01 = xt + ((b * Hh + y0c) * Ww + x1c) * Cin;
    const float* p10 = xt + ((b * Hh + y1c) * Ww + x0c) * Cin;
    const float* p11 = xt + ((b * Hh + y1c) * Ww + x1c) * Cin;
    float* bcol = Bs + (k * 64) * 32 + n;
#pragma unroll 4
    for (int c4 = 0; c4 < 16; ++c4) {
      float4 a00 = *(const float4*)(p00 + c4 * 4);
      float4 a01 = *(const float4*)(p01 + c4 * 4);
      float4 a10 = *(const float4*)(p10 + c4 * 4);
      float4 a11 = *(const float4*)(p11 + c4 * 4);
      bcol[(c4 * 4 + 0) * 32] = w00 * a00.x + w01 * a01.x + w10 * a10.x + w11 * a11.x;
      bcol[(c4 * 4 + 1) * 32] = w00 * a00.y + w01 * a01.y + w10 * a10.y + w11 * a11.y;
      bcol[(c4 * 4 + 2) * 32] = w00 * a00.z + w01 * a01.z + w10 * a10.z + w11 * a11.z;
      bcol[(c4 * 4 + 3) * 32] = w00 * a00.w + w01 * a01.w + w10 * a10.w + w11 * a11.w;
    }
  }
  __syncthreads();

  // --- WMMA: wave -> (mt, nt) tile; ISA f32 layouts:
  // A 16x4: lanes 0-15 hold K=r0,r0+1; lanes 16-31 hold K=r0+2,r0+3.
  // B 4x16: VGPR0 rows r0 / r0+2 split by lane half, VGPR1 rows r0+1 / r0+3.
  int wave = tid >> 5, lane = tid & 31;
  int mt = wave >> 1, nt = wave & 1;
  int mbase = mt * 16, nbase = nt * 16;
  int lhalf = lane >> 4, l16 = lane & 15;
  const float* Arow = dwr + (mbase + l16) * RED + lhalf * 2;  // L2-hot weights
  const float* Brow = Bs + (lhalf * 2) * 32 + nbase + l16;
  v8f acc = {};
#pragma unroll 4
  for (int kk = 0; kk < RED / 4; ++kk) {
    int r0 = kk * 4;
    v2f a = *(const v2f*)(Arow + r0);          // 8B aligned (RED even, r0 mult 4)
    v2f bb;
    bb.x = Brow[r0 * 32];
    bb.y = Brow[(r0 + 1) * 32];
    acc = __builtin_amdgcn_wmma_f32_16x16x4_f32(false, a, false, bb,
                                                (short)0, acc, false, false);
  }

  // --- store C tile (NCHW) + bias. VGPR i holds rows i / i+8 by lane half.
  int o_lo = mbase + lhalf * 8;
  int w = w0 + nbase + l16;
#pragma unroll
  for (int i = 0; i < 8; ++i) {
    int o = o_lo + i;
    out[((b * Cout + o) * Hh + h) * Ww + w] = acc[i] + db[o];
  }
}

}  // namespace

extern "C" void kernel_launch(void* const* d_in, const int* in_sizes, int n_in,
                              void* d_out, int out_size, void* d_ws, size_t ws_size,
                              hipStream_t stream) {
  const float* x  = (const float*)d_in[0];
  const float* ow = (const float*)d_in[1];
  const float* ob = (const float*)d_in[2];
  const float* dw = (const float*)d_in[3];
  const float* db = (const float*)d_in[4];
  float* out = (float*)d_out;

  // Workspace layout: xt (16 MB) | offs (4.5 MB) | dwr (144 KB)  ~= 21 MB
  float* xt   = (float*)d_ws;
  float* offs = xt + (size_t)Bn * NPIX * Cin;
  float* dwr  = offs + (size_t)Bn * NPIX * 18;

  k_transpose <<<dim3(8, Hh, Bn), 256, 0, stream>>>(x, xt);
  k_reorder_w <<<dim3((Cout * RED + 255) / 256), 256, 0, stream>>>(dw, dwr);
  k_offset_conv<<<dim3(64, Bn), 256, 0, stream>>>(xt, ow, ob, offs);
  k_deform_wmma<<<dim3((Ww / 32) * Hh, Bn), 256, 0, stream>>>(xt, offs, dwr, db, out);
}